// Network_781684048177
// MI455X (gfx1250) — compile-verified
//
#include <hip/hip_runtime.h>

// ---------------------------------------------------------------------------
// LIF + STDP recurrent network, T=64 steps, N=2048 neurons, W is N x N f32.
//
// Perf model (MI455X): W = 16MB fits in 192MB L2, so each step is pure L2
// streaming: one fused pass reads W, applies the STDP update in registers,
// writes W back, and accumulates the next step's matvec with
// v_wmma_f32_16x16x4_f32 (W tile lives directly in the WMMA A-layout).
// The t-loop is sequential; per step:
//   neuron_step : O(N) LIF/trace update, sums 4 i_syn partials   (tiny)
//   stdp_matvec : 128 row-tiles x 4 K-splits = 512 wave32 workgroups,
//                 each streaming a 16x512 panel (32KB rd + 32KB wr) --
//                 enough waves in flight to saturate L2 bandwidth.
// K-split partials go to disjoint buffers (no float atomics) so results are
// bitwise deterministic across graph replays.
// ---------------------------------------------------------------------------

#define T_STEPS 64
#define NPOP    2048
#define KSPLIT  4
#define KCHUNK  (NPOP / KSPLIT)   // 512 columns per wave

__device__ __constant__ float kDT           = 1e-3f;
__device__ __constant__ float kTAU_MEM_INV  = 100.0f;
__device__ __constant__ float kV_TH         = 1.0f;
__device__ __constant__ float kTAU_PRE_INV  = 50.0f;
__device__ __constant__ float kTAU_POST_INV = 50.0f;
__device__ __constant__ float kETA_PLUS  = 1e-3f;
__device__ __constant__ float kETA_MINUS = 1e-3f;
__device__ __constant__ float kW_MIN = 0.0f;
__device__ __constant__ float kW_MAX = 1.0f;

typedef __attribute__((ext_vector_type(2))) float v2f;
typedef __attribute__((ext_vector_type(8))) float v8f;

// ---------------------------------------------------------------------------
// One-time state init: v = z = isyn_part = 0, traces copied from inputs.
// ---------------------------------------------------------------------------
__global__ void init_state(const float* __restrict__ t_pre,
                           const float* __restrict__ t_post,
                           float* __restrict__ v,   float* __restrict__ tp,
                           float* __restrict__ tpo, float* __restrict__ z,
                           float* __restrict__ isyn) {
    int i = blockIdx.x * blockDim.x + threadIdx.x;
    if (i < NPOP) {
        v[i] = 0.0f; z[i] = 0.0f;
        tp[i]  = t_pre[i];
        tpo[i] = t_post[i];
        #pragma unroll
        for (int p = 0; p < KSPLIT; ++p) isyn[p * NPOP + i] = 0.0f;
    }
}

// ---------------------------------------------------------------------------
// Per-step neuron update: membrane, spike (Heaviside fwd), reset, traces.
// i_syn = sum of the KSPLIT partials from the previous fused kernel.
// ---------------------------------------------------------------------------
__global__ void neuron_step(const float* __restrict__ x,      // exc_current row t
                            float* __restrict__ v,
                            float* __restrict__ tp,
                            float* __restrict__ tpo,
                            float* __restrict__ z,
                            const float* __restrict__ isyn,   // [KSPLIT][N]
                            float* __restrict__ out_row) {    // spikes row t
    int i = blockIdx.x * blockDim.x + threadIdx.x;
    if (i >= NPOP) return;
    float isyn_i = isyn[i] + isyn[NPOP + i] + isyn[2 * NPOP + i] + isyn[3 * NPOP + i];
    float vi = v[i];
    vi = vi + kDT * kTAU_MEM_INV * ((0.0f - vi) + isyn_i + x[i]);
    float zi = (vi - kV_TH) > 0.0f ? 1.0f : 0.0f;
    vi = vi * (1.0f - zi);                       // reset to V_RESET = 0
    v[i] = vi;
    float tpi = tp[i];
    tpi += kDT * kTAU_PRE_INV * (-tpi + zi);     // A_PRE = 1
    tp[i] = tpi;
    float tpoi = tpo[i];
    tpoi += kDT * kTAU_POST_INV * (-tpoi + zi);  // A_POST = 1
    tpo[i] = tpoi;
    z[i] = zi;
    out_row[i] = zi;
}

// ---------------------------------------------------------------------------
// Fused STDP weight update (step t) + partial matvec for step t+1:
//   W <- clip(W + etaP*outer(z,tp) - etaM*outer(tpo,z))   [z = z_t, traces_t]
//   isyn_part[ks] <- W_new[:, Krange] @ z[Krange]
// One wave per (16-row tile, K-split). W elements live in the WMMA f32
// A-matrix layout: lane (l&15) = row m, (l>>4)*2 = K offset; VGPR0/1 = K,K+1.
// B = z[k] broadcast across all 16 N-columns (columns identical -> N layout
// irrelevant; K mapping mirrors A's). D column 0 = partial i_syn for the tile.
// ---------------------------------------------------------------------------
__launch_bounds__(32)
__global__ void stdp_matvec(float* __restrict__ w,
                            const float* __restrict__ z,
                            const float* __restrict__ tp,
                            const float* __restrict__ tpo,
                            float* __restrict__ isyn) {       // [KSPLIT][N]
    __shared__ float z_s[KCHUNK];
    __shared__ float tp_s[KCHUNK];

    const int lane  = threadIdx.x;            // 0..31, one wave32
    const int row0  = blockIdx.x * 16;        // 16-row tile of W
    const int kbase = blockIdx.y * KCHUNK;    // this wave's column range

    // Stage this K-slice of the broadcast operands into LDS (float4 loads).
    const float4* zv = (const float4*)(z + kbase);
    const float4* tv = (const float4*)(tp + kbase);
    float4* zs4 = (float4*)z_s;
    float4* ts4 = (float4*)tp_s;
    for (int i = lane; i < KCHUNK / 4; i += 32) {
        zs4[i] = zv[i];
        ts4[i] = tv[i];
    }
    __syncthreads();

    const int   m      = lane & 15;           // row within tile (A layout)
    const int   khi    = (lane >> 4) * 2;     // K offset for this half-wave
    const int   row    = row0 + m;
    const float zrow   = z[row];              // z_new[post]  (L2-hot broadcast)
    const float tporow = tpo[row];            // tpo[post]
    float* __restrict__ wrow = w + (size_t)row * NPOP + kbase;

    v8f c = {};                               // partial i_syn accumulator
    for (int k0 = 0; k0 < KCHUNK; k0 += 4) {
        const int k = k0 + khi;
        v2f a = *(const v2f*)(wrow + k);      // w[row, kbase+k], w[row, kbase+k+1]

        // additive STDP with hard bounds
        float a0 = a.x + kETA_PLUS * zrow * tp_s[k]     - kETA_MINUS * tporow * z_s[k];
        float a1 = a.y + kETA_PLUS * zrow * tp_s[k + 1] - kETA_MINUS * tporow * z_s[k + 1];
        a.x = fminf(fmaxf(a0, kW_MIN), kW_MAX);
        a.y = fminf(fmaxf(a1, kW_MIN), kW_MAX);
        *(v2f*)(wrow + k) = a;                // persist W^{t+1}

        // B[k, n] = z[kbase + k0 + k] for every column n (matvec RHS)
        v2f b;
        b.x = z_s[k];
        b.y = z_s[k + 1];

        // D += A(16x4) * B(4x16); column 0 accumulates W_new @ z for the tile
        c = __builtin_amdgcn_wmma_f32_16x16x4_f32(
                /*neg_a=*/false, a, /*neg_b=*/false, b,
                /*c_mod=*/(short)0, c, /*reuse_a=*/false, /*reuse_b=*/false);
    }

    // D layout: N=0 column lives in lane 0 (M=0..7) and lane 16 (M=8..15).
    float* part = isyn + (size_t)blockIdx.y * NPOP;
    if (lane == 0) {
        #pragma unroll
        for (int mm = 0; mm < 8; ++mm) part[row0 + mm] = c[mm];
    } else if (lane == 16) {
        #pragma unroll
        for (int mm = 0; mm < 8; ++mm) part[row0 + 8 + mm] = c[mm];
    }
}

// ---------------------------------------------------------------------------
// Launch: 1 init + T neuron kernels + (T-1) fused kernels (last W update
// cannot influence the spike output, so it is skipped).
// Workspace (f32): [N*N] W copy | [N] v | [N] tp | [N] tpo | [N] z |
//                  [KSPLIT*N] isyn partials.   ws_size >= ~16.85 MB.
// ---------------------------------------------------------------------------
extern "C" void kernel_launch(void* const* d_in, const int* in_sizes, int n_in,
                              void* d_out, int out_size, void* d_ws, size_t ws_size,
                              hipStream_t stream) {
    const float* exc    = (const float*)d_in[0];   // [T, N]
    const float* w_in   = (const float*)d_in[1];   // [N, N]
    const float* t_pre  = (const float*)d_in[2];   // [N]
    const float* t_post = (const float*)d_in[3];   // [N]
    float*       out    = (float*)d_out;           // [T, N] spikes

    float* w    = (float*)d_ws;
    float* v    = w + (size_t)NPOP * NPOP;
    float* tp   = v + NPOP;
    float* tpo  = tp + NPOP;
    float* z    = tpo + NPOP;
    float* isyn = z + NPOP;                        // KSPLIT * N floats

    // Fresh W copy every call (inputs must not be mutated; deterministic).
    hipMemcpyAsync(w, w_in, sizeof(float) * (size_t)NPOP * NPOP,
                   hipMemcpyDeviceToDevice, stream);
    init_state<<<(NPOP + 255) / 256, 256, 0, stream>>>(t_pre, t_post, v, tp, tpo, z, isyn);

    dim3 fgrid(NPOP / 16, KSPLIT);                 // 128 tiles x 4 K-splits
    for (int t = 0; t < T_STEPS; ++t) {
        neuron_step<<<NPOP / 256, 256, 0, stream>>>(
            exc + (size_t)t * NPOP, v, tp, tpo, z, isyn, out + (size_t)t * NPOP);
        if (t + 1 < T_STEPS) {
            stdp_matvec<<<fgrid, 32, 0, stream>>>(w, z, tp, tpo, isyn);
        }
    }
}